// GCN_37434934952809
// MI455X (gfx1250) — compile-verified
//
#include <hip/hip_runtime.h>

#define N_NODES   100000
#define N_EDGES   1600000
#define N_GRAPHS  1000
#define DIN       128
#define DMID      128
#define DGLOB     64
#define DOUT      2

typedef __attribute__((ext_vector_type(2))) float v2f;
typedef __attribute__((ext_vector_type(8))) float v8f;

__device__ __forceinline__ void atomAddF32(float* p, float v) {
    __hip_atomic_fetch_add(p, v, __ATOMIC_RELAXED, __HIP_MEMORY_SCOPE_AGENT);
}

// ---------- small elementwise kernels ----------
__global__ void fillK(float* __restrict__ p, float v, int n) {
    int i = blockIdx.x * blockDim.x + threadIdx.x;
    if (i < n) p[i] = v;
}

__global__ void degScatterK(const int* __restrict__ col, float* __restrict__ deg) {
    int e = blockIdx.x * blockDim.x + threadIdx.x;
    if (e < N_EDGES) atomAddF32(&deg[col[e]], 1.0f);
}

__global__ void rsqrtK(float* __restrict__ d, int n) {
    int i = blockIdx.x * blockDim.x + threadIdx.x;
    if (i < n) d[i] = __frsqrt_rn(d[i]);   // deg >= 1 always (self loops)
}

// Pack W (K=128 x N=128, row-major) into WMMA B-fragment order:
// for k-step ks (0..31), n-tile nt (0..7), lane l (0..31), comp c (0..1):
//   k = ks*4 + (l>=16 ? 2:0) + c ; n = nt*16 + (l&15)
__global__ void packWK(const float* __restrict__ W, float* __restrict__ Wp) {
    int tid = blockIdx.x * blockDim.x + threadIdx.x;   // 0 .. 16383
    if (tid >= 128 * 128) return;
    int c    = tid & 1;
    int lane = (tid >> 1) & 31;
    int nt   = (tid >> 6) & 7;
    int ks   = tid >> 9;
    int k = ks * 4 + ((lane & 16) ? 2 : 0) + c;
    int n = nt * 16 + (lane & 15);
    Wp[tid] = W[k * 128 + n];
}

// ---------- WMMA fp32 GEMM:  t[nRows x 128] = act(A) @ W ----------
// One wave -> 16 rows x 128 cols of output. FUSE: A-element = relu(A + bias[k]).
// Fused epilogue: also writes agg[r,:] = dinv[r]^2 * t[r,:]  (self-loop init).
template <bool FUSE>
__global__ void gemmK(const float* __restrict__ A, const float* __restrict__ Wp,
                      const float* __restrict__ bias, const float* __restrict__ dinv,
                      float* __restrict__ outT, float* __restrict__ outAgg,
                      int nRows) {
    const int lane    = threadIdx.x & 31;
    const int wave    = threadIdx.x >> 5;
    const int rowBase = blockIdx.x * 128 + wave * 16;   // multiple of 16
    if (rowBase >= nRows) return;                        // wave-uniform: EXEC stays all-1
    const int half = lane >> 4;        // 0 | 1
    const int l16  = lane & 15;
    const int koff = half * 2;
    const float* aRow = A + (size_t)(rowBase + l16) * 128 + koff;
    const v2f*  wp    = (const v2f*)Wp;

    v8f acc[8] = {};
#pragma unroll 4
    for (int ks = 0; ks < 32; ++ks) {
        v2f a = *(const v2f*)(aRow + ks * 4);
        if (FUSE) {
            const int k = ks * 4 + koff;
            a.x = fmaxf(a.x + bias[k], 0.0f);
            a.y = fmaxf(a.y + bias[k + 1], 0.0f);
        }
#pragma unroll
        for (int nt = 0; nt < 8; ++nt) {
            v2f b = wp[(ks * 8 + nt) * 32 + lane];
            acc[nt] = __builtin_amdgcn_wmma_f32_16x16x4_f32(
                false, a, false, b, (short)0, acc[nt], false, false);
        }
    }
    // D layout: VGPR v, lanes 0-15 -> M=v, lanes 16-31 -> M=v+8, N=lane&15
    float s2[8];
#pragma unroll
    for (int v = 0; v < 8; ++v) {
        float dv = dinv[rowBase + v + half * 8];
        s2[v] = dv * dv;
    }
#pragma unroll
    for (int nt = 0; nt < 8; ++nt) {
#pragma unroll
        for (int v = 0; v < 8; ++v) {
            const size_t idx = (size_t)(rowBase + v + half * 8) * 128 + nt * 16 + l16;
            const float val = acc[nt][v];
            outT[idx]   = val;
            outAgg[idx] = s2[v] * val;   // self-loop contribution, inits accumulator
        }
    }
}

// agg[col] += dinv[row]*dinv[col] * t[row] : one wave per edge, float4 per lane
__global__ void scatterK(const float* __restrict__ t, const float* __restrict__ dinv,
                         const int* __restrict__ row, const int* __restrict__ col,
                         float* __restrict__ agg) {
    int gid = blockIdx.x * blockDim.x + threadIdx.x;
    int e = gid >> 5, lane = gid & 31;
    if (e >= N_EDGES) return;
    int r = row[e], c = col[e];
    float nrm = dinv[r] * dinv[c];
    const float4 v = *(const float4*)(t + (size_t)r * 128 + lane * 4);
    float* d = agg + (size_t)c * 128 + lane * 4;
    atomAddF32(d + 0, nrm * v.x);
    atomAddF32(d + 1, nrm * v.y);
    atomAddF32(d + 2, nrm * v.z);
    atomAddF32(d + 3, nrm * v.w);
}

// sums[batch[i]] += relu(agg[i]+b2); cnt[batch[i]] += 1 : one wave per node
__global__ void poolK(const float* __restrict__ agg, const float* __restrict__ b2,
                      const int* __restrict__ batch, float* __restrict__ sums,
                      float* __restrict__ cnt) {
    int gid = blockIdx.x * blockDim.x + threadIdx.x;
    int node = gid >> 5, lane = gid & 31;
    if (node >= N_NODES) return;
    int g = batch[node];
    const float4 v = *(const float4*)(agg + (size_t)node * 128 + lane * 4);
    const float4 bb = *(const float4*)(b2 + lane * 4);
    float* d = sums + (size_t)g * 128 + lane * 4;
    atomAddF32(d + 0, fmaxf(v.x + bb.x, 0.0f));
    atomAddF32(d + 1, fmaxf(v.y + bb.y, 0.0f));
    atomAddF32(d + 2, fmaxf(v.z + bb.z, 0.0f));
    atomAddF32(d + 3, fmaxf(v.w + bb.w, 0.0f));
    if (lane == 0) atomAddF32(&cnt[g], 1.0f);
}

// head MLP: out[g] = relu([sums[g]/cnt[g], u[g]] @ Wh1 + bh1) @ Wh2 + bh2
__global__ void headK(const float* __restrict__ sums, const float* __restrict__ cnt,
                      const float* __restrict__ u, const float* __restrict__ Wh1,
                      const float* __restrict__ bh1, const float* __restrict__ Wh2,
                      const float* __restrict__ bh2, float* __restrict__ out) {
    const int g = blockIdx.x, t = threadIdx.x;   // 128 threads
    __shared__ float emb[DMID + DGLOB];
    __shared__ float red[128][2];
    float c = fmaxf(cnt[g], 1.0f);
    emb[t] = sums[(size_t)g * 128 + t] / c;
    if (t < DGLOB) emb[DMID + t] = u[(size_t)g * DGLOB + t];
    __syncthreads();
    float h = bh1[t];
#pragma unroll 8
    for (int j = 0; j < DMID + DGLOB; ++j) h = fmaf(emb[j], Wh1[j * 128 + t], h);
    h = fmaxf(h, 0.0f);
    red[t][0] = h * Wh2[t * 2 + 0];
    red[t][1] = h * Wh2[t * 2 + 1];
    __syncthreads();
    for (int s = 64; s > 0; s >>= 1) {
        if (t < s) { red[t][0] += red[t + s][0]; red[t][1] += red[t + s][1]; }
        __syncthreads();
    }
    if (t == 0) {
        out[g * 2 + 0] = red[0][0] + bh2[0];
        out[g * 2 + 1] = red[0][1] + bh2[1];
    }
}

extern "C" void kernel_launch(void* const* d_in, const int* in_sizes, int n_in,
                              void* d_out, int out_size, void* d_ws, size_t ws_size,
                              hipStream_t stream) {
    const float* x     = (const float*)d_in[0];
    const int*   ei    = (const int*)d_in[1];
    const int*   row   = ei;
    const int*   col   = ei + N_EDGES;
    const float* u     = (const float*)d_in[2];
    const int*   batch = (const int*)d_in[3];
    const float* W1  = (const float*)d_in[5];
    const float* b1  = (const float*)d_in[6];
    const float* W2  = (const float*)d_in[7];
    const float* b2  = (const float*)d_in[8];
    const float* Wh1 = (const float*)d_in[9];
    const float* bh1 = (const float*)d_in[10];
    const float* Wh2 = (const float*)d_in[11];
    const float* bh2 = (const float*)d_in[12];
    float* out = (float*)d_out;

    // workspace layout (bytes, 4KB-aligned chunks)
    char* ws = (char*)d_ws;
    float* dinv  = (float*)(ws + 0);                         // 400,000 B
    float* bufT  = (float*)(ws + (size_t)524288);            // 51,200,000 B (t1/t2)
    float* bufAg = (float*)(ws + (size_t)51724288);          // 51,200,000 B (agg1/agg2)
    float* sums  = (float*)(ws + (size_t)102924288);         // 512,000 B
    float* cnt   = sums + N_GRAPHS * 128;                    // 4,000 B (contiguous)
    float* wpack = (float*)(ws + (size_t)103440384);         // 65,536 B

    const dim3 B256(256);
    // 1) degrees (init to 1 for self loops) -> dinv
    fillK<<<(N_NODES + 255) / 256, B256, 0, stream>>>(dinv, 1.0f, N_NODES);
    degScatterK<<<(N_EDGES + 255) / 256, B256, 0, stream>>>(col, dinv);
    rsqrtK<<<(N_NODES + 255) / 256, B256, 0, stream>>>(dinv, N_NODES);

    // 2) layer 1: t1 = x @ W1 ; agg1 = dinv^2*t1 (fused) ; + edge scatter
    packWK<<<64, B256, 0, stream>>>(W1, wpack);
    gemmK<false><<<(N_NODES + 127) / 128, B256, 0, stream>>>(x, wpack, nullptr, dinv,
                                                             bufT, bufAg, N_NODES);
    scatterK<<<(N_EDGES * 32 + 255) / 256, B256, 0, stream>>>(bufT, dinv, row, col, bufAg);

    // 3) layer 2: t2 = relu(agg1 + b1) @ W2 ; agg2 = dinv^2*t2 (fused) ; + edge scatter
    packWK<<<64, B256, 0, stream>>>(W2, wpack);
    gemmK<true><<<(N_NODES + 127) / 128, B256, 0, stream>>>(bufAg, wpack, b1, dinv,
                                                            bufT, bufAg, N_NODES);
    scatterK<<<(N_EDGES * 32 + 255) / 256, B256, 0, stream>>>(bufT, dinv, row, col, bufAg);

    // 4) mean pool (h2 = relu(agg2 + b2) fused) and head MLP
    fillK<<<(N_GRAPHS * 129 + 255) / 256, B256, 0, stream>>>(sums, 0.0f, N_GRAPHS * 129);
    poolK<<<(N_NODES * 32 + 255) / 256, B256, 0, stream>>>(bufAg, b2, batch, sums, cnt);
    headK<<<N_GRAPHS, dim3(128), 0, stream>>>(sums, cnt, u, Wh1, bh1, Wh2, bh2, out);
}